// _Shiftnd_51591147159939
// MI455X (gfx1250) — compile-verified
//
#include <hip/hip_runtime.h>

#define HH 128
#define WW 128
#define PLANE (HH * WW)
#define TPB 256

#if defined(__has_builtin)
#  if __has_builtin(__builtin_amdgcn_global_load_async_to_lds_b128)
#    define HAVE_ASYNC_B128 1
#  endif
#  if __has_builtin(__builtin_amdgcn_s_wait_asynccnt)
#    define HAVE_WAIT_ASYNC 1
#  endif
#endif

// Builtin parameter type (from hipcc diagnostic):
//   'int __attribute__((__vector_size__(4 * sizeof(int)))) __device__ *'
typedef int v4i __attribute__((vector_size(16)));
typedef __attribute__((address_space(1))) v4i gv4i;   // global  (__device__)
typedef __attribute__((address_space(3))) v4i sv4i;   // LDS     (__shared__)

typedef float f4 __attribute__((vector_size(16)));

__global__ __launch_bounds__(TPB) void shift2d_kernel(
    const float* __restrict__ x, const float* __restrict__ wt, int nchan,
    float* __restrict__ y)
{
    __shared__ float tile[PLANE];              // one 64KB plane per workgroup

    const int pi  = blockIdx.x;                // plane index = b*nchan + c
    const int c   = pi % nchan;
    const long long base = (long long)pi * PLANE;
    const int tid = threadIdx.x;

    // ---------------- stage whole plane into LDS (async b128, coalesced) --------
#if defined(HAVE_ASYNC_B128)
    {
        gv4i* g = (gv4i*)(x + base);
        sv4i* l = (sv4i*)tile;
#pragma unroll
        for (int k = 0; k < PLANE / 4 / TPB; ++k) {
            int i = tid + k * TPB;
            __builtin_amdgcn_global_load_async_to_lds_b128(g + i, l + i, 0, 0);
        }
    }
#  if defined(HAVE_WAIT_ASYNC)
    __builtin_amdgcn_s_wait_asynccnt(0);
#  else
    asm volatile("s_wait_asynccnt 0" ::: "memory");
#  endif
#else
    {
        const f4* gsrc = (const f4*)(x + base);
        f4* l = (f4*)tile;
#pragma unroll
        for (int k = 0; k < PLANE / 4 / TPB; ++k) {
            int i = tid + k * TPB;
            l[i] = gsrc[i];
        }
    }
#endif
    __syncthreads();

    // ---------------- per-channel constants (uniform across the block) ----------
    const float sh = wt[2 * c + 0];
    const float sw = wt[2 * c + 1];
    const float fhf = floorf(sh), fwf = floorf(sw);
    const int   s0 = (int)fhf,   t0 = (int)fwf;
    const float fh = sh - fhf,   fw = sw - fwf;
    const float c00 = (1.0f - fh) * (1.0f - fw);
    const float c01 = (1.0f - fh) * fw;
    const float c10 = fh * (1.0f - fw);
    const float c11 = fh * fw;

    // ---------------- 4-tap stencil with row-recurrence --------------------------
    // Thread (half, w) walks h = half*64 + k. Bottom-row taps of pixel (h,w)
    // become the top-row taps of pixel (h+1,w): 2 LDS loads per pixel instead of 4.
    const int w       = tid & (WW - 1);        // lanes stride-1 in w: conflict-free
    const int half    = tid >> 7;              // 0 or 1
    const int h_begin = half * (HH / 2);

    const int  w0c = w + t0, w1c = w0c + 1;    // loop-invariant column taps
    const bool vw0 = (unsigned)w0c < (unsigned)WW;
    const bool vw1 = (unsigned)w1c < (unsigned)WW;
    const int  q0  = vw0 ? w0c : 0;
    const int  q1  = vw1 ? w1c : 0;

    // prime the top-row taps: row h_begin + s0
    {
        const int  ht = h_begin + s0;
        const bool vt = (unsigned)ht < (unsigned)HH;
        const int  rt = (vt ? ht : 0) * WW;
        float top0 = (vt && vw0) ? tile[rt + q0] : 0.0f;
        float top1 = (vt && vw1) ? tile[rt + q1] : 0.0f;

        float* yrow = y + base + (long long)h_begin * WW + w;
#pragma unroll 4
        for (int k = 0; k < HH / 2; ++k) {
            const int  hb = h_begin + k + s0 + 1;        // bottom tap row
            const bool vb = (unsigned)hb < (unsigned)HH;
            const int  rb = (vb ? hb : 0) * WW;
            const float bot0 = (vb && vw0) ? tile[rb + q0] : 0.0f;
            const float bot1 = (vb && vw1) ? tile[rb + q1] : 0.0f;

            yrow[k * WW] = c00 * top0 + c01 * top1 + c10 * bot0 + c11 * bot1;

            top0 = bot0;                                  // row recurrence
            top1 = bot1;
        }
    }
}

__global__ __launch_bounds__(TPB) void l1_loss_kernel(
    const float* __restrict__ w, int n, float* __restrict__ out)
{
    __shared__ float red[TPB];
    float s = 0.0f;
    for (int i = threadIdx.x; i < n; i += TPB) s += fabsf(w[i]);
    red[threadIdx.x] = s;
    __syncthreads();
    for (int off = TPB / 2; off > 0; off >>= 1) {
        if (threadIdx.x < off) red[threadIdx.x] += red[threadIdx.x + off];
        __syncthreads();
    }
    if (threadIdx.x == 0) out[0] = 0.0005f * red[0];
}

extern "C" void kernel_launch(void* const* d_in, const int* in_sizes, int n_in,
                              void* d_out, int out_size, void* d_ws, size_t ws_size,
                              hipStream_t stream) {
    const float* x  = (const float*)d_in[0];   // [16,256,128,128] f32
    const float* wt = (const float*)d_in[1];   // [256,2] f32
    float* y = (float*)d_out;                  // y flat, then loss scalar

    const int nx     = in_sizes[0];            // 16*256*128*128
    const int nw     = in_sizes[1];            // 512
    const int nchan  = nw / 2;                 // 256
    const int planes = nx / PLANE;             // 4096

    shift2d_kernel<<<planes, TPB, 0, stream>>>(x, wt, nchan, y);
    l1_loss_kernel<<<1, TPB, 0, stream>>>(wt, nw, y + (long long)nx);
}